// Yolov1Loss_43001212567716
// MI455X (gfx1250) — compile-verified
//
#include <hip/hip_runtime.h>

#define GS 7
#define NB 2
#define NC 20
#define CH 30                       // 5*NB + NC
#define BATCH 8192
#define NCELLS (BATCH * GS * GS)    // 401408
#define TPB 256
#define CELLS_PER_BLOCK 256
#define TILE_FLOATS (CELLS_PER_BLOCK * CH)   // 7680 floats
#define TILE_BYTES (TILE_FLOATS * 4)         // 30720 bytes
#define CHUNKS (TILE_BYTES / 16)             // 1920 x b128 per tensor
#define FULL_ROUNDS (CHUNKS / TPB)           // 7
#define TAIL (CHUNKS - FULL_ROUNDS * TPB)    // 128
#define NBLOCKS (NCELLS / CELLS_PER_BLOCK)   // 1568

// ---- CDNA5 async global->LDS path (guarded) --------------------------------
typedef int v4i __attribute__((ext_vector_type(4)));
typedef __attribute__((address_space(1))) v4i gv4i_t;   // global b128
typedef __attribute__((address_space(3))) v4i lv4i_t;   // LDS b128

#if defined(__has_builtin)
#if __has_builtin(__builtin_amdgcn_global_load_async_to_lds_b128)
#define HAVE_ASYNC_LDS 1
#endif
#if __has_builtin(__builtin_amdgcn_s_wait_asynccnt)
#define HAVE_WAIT_ASYNC 1
#endif
#endif

__device__ __forceinline__ void wait_async0() {
#if defined(HAVE_WAIT_ASYNC)
    __builtin_amdgcn_s_wait_asynccnt(0);
#else
    asm volatile("s_wait_asynccnt 0" ::: "memory");
#endif
}

// One b128 async copy; INST_OFFSET applies to BOTH the global and LDS address
// (CDNA5 ISA), so a single base pair serves all rounds.
template <int OFF>
__device__ __forceinline__ void async_cp16(const char* g, char* l) {
#if defined(HAVE_ASYNC_LDS)
    __builtin_amdgcn_global_load_async_to_lds_b128((gv4i_t*)g, (lv4i_t*)l, OFF, 0);
#else
    *(int4*)(l + OFF) = *(const int4*)(g + OFF);
#endif
}

__device__ __forceinline__ void stage_tensor(const char* g, char* l, int tid) {
    // 7 full rounds of 256 chunks each: literal offsets, no address math.
    async_cp16<0 * TPB * 16>(g, l);
    async_cp16<1 * TPB * 16>(g, l);
    async_cp16<2 * TPB * 16>(g, l);
    async_cp16<3 * TPB * 16>(g, l);
    async_cp16<4 * TPB * 16>(g, l);
    async_cp16<5 * TPB * 16>(g, l);
    async_cp16<6 * TPB * 16>(g, l);
    // half-round tail: first 128 threads only
    if (tid < TAIL) async_cp16<FULL_ROUNDS * TPB * 16>(g, l);
}

// ---------------------------------------------------------------------------
// Kernel 1: each block stages one 256-cell tile of inputs+targets into LDS via
// async b128 copies, computes per-cell YOLO loss terms, and writes 3 partial
// sums per block.
__global__ __launch_bounds__(TPB) void yolo_partial_kernel(
    const float* __restrict__ in, const float* __restrict__ tg,
    float* __restrict__ partial) {
    __shared__ float lds[2 * TILE_FLOATS];   // [0..7679]=inputs, [7680..]=targets
    __shared__ float wsum[TPB / 32][3];

    const int tid = threadIdx.x;
    const size_t tile_off = (size_t)blockIdx.x * TILE_BYTES + (size_t)tid * 16;
    char* lds_b = (char*)lds + (size_t)tid * 16;

    // ---- stage 2 x 30720 B into LDS (async DMA, ASYNCcnt-tracked) ----------
    stage_tensor((const char*)in + tile_off, lds_b, tid);
    stage_tensor((const char*)tg + tile_off, lds_b + TILE_BYTES, tid);
#if defined(HAVE_ASYNC_LDS)
    wait_async0();
#endif
    __syncthreads();

    // ---- per-cell loss (one thread = one grid cell) ------------------------
    const float* ci = lds + tid * CH;                // this cell's predictions
    const float* ct = lds + TILE_FLOATS + tid * CH;  // this cell's targets
    const float inv_gs = 1.0f / (float)GS;

    const float tconf = ct[4];
    const float coord = (tconf > 0.0f) ? 1.0f : 0.0f;
    const float noobj = (tconf == 0.0f) ? 1.0f : 0.0f;

    // no-object confidence loss (both boxes, conf channels 4 and 9)
    const float dn0 = ci[4] - ct[4];
    const float dn1 = ci[9] - ct[9];
    const float l_noobj = noobj * (dn0 * dn0 + dn1 * dn1);

    // target box in xyxy
    const float tx = ct[0] * inv_gs, ty = ct[1] * inv_gs;
    const float tw = ct[2], th = ct[3];
    const float tx1 = tx - 0.5f * tw, ty1 = ty - 0.5f * th;
    const float tx2 = tx + 0.5f * tw, ty2 = ty + 0.5f * th;
    const float t_area = tw * th;

    float iou0 = 0.0f, iou1 = 0.0f;
#pragma unroll
    for (int b = 0; b < NB; ++b) {
        const float px = ci[5 * b + 0] * inv_gs, py = ci[5 * b + 1] * inv_gs;
        const float pw = ci[5 * b + 2], ph = ci[5 * b + 3];
        const float px1 = px - 0.5f * pw, py1 = py - 0.5f * ph;
        const float px2 = px + 0.5f * pw, py2 = py + 0.5f * ph;
        const float ltx = fmaxf(px1, tx1), lty = fmaxf(py1, ty1);
        const float rbx = fminf(px2, tx2), rby = fminf(py2, ty2);
        const float wx = fmaxf(rbx - ltx, 0.0f), wy = fmaxf(rby - lty, 0.0f);
        const float inter = wx * wy;
        const float v = inter / (pw * ph + t_area - inter);
        if (b == 0) iou0 = v; else iou1 = v;
    }
    const int idx = (iou1 > iou0) ? 1 : 0;          // jnp.argmax: first max wins
    const float max_iou = fmaxf(iou0, iou1);

    const float* rp = ci + 5 * idx;                 // responsible box
    const float dx = rp[0] - ct[0], dy = rp[1] - ct[1];
    const float l_xy = dx * dx + dy * dy;
    const float sw = sqrtf(rp[2]) - sqrtf(ct[2]);
    const float sh = sqrtf(rp[3]) - sqrtf(ct[3]);
    const float l_wh = sw * sw + sh * sh;
    const float dob = rp[4] - max_iou;
    const float l_obj = dob * dob;

    float l_class = 0.0f;
#pragma unroll
    for (int c = 5 * NB; c < CH; ++c) {
        const float d = ci[c] - ct[c];
        l_class += d * d;
    }

    // fold lambdas: boxes = 0.5*(xy+wh), conf = obj + 0.5*noobj
    float p_boxes = coord * 0.5f * (l_xy + l_wh);
    float p_conf  = coord * l_obj + 0.5f * l_noobj;
    float p_class = coord * l_class;

    // ---- wave32 tree reduction --------------------------------------------
#pragma unroll
    for (int off = 16; off >= 1; off >>= 1) {
        p_boxes += __shfl_xor(p_boxes, off, 32);
        p_conf  += __shfl_xor(p_conf,  off, 32);
        p_class += __shfl_xor(p_class, off, 32);
    }
    const int wave = tid >> 5;
    if ((tid & 31) == 0) {
        wsum[wave][0] = p_boxes;
        wsum[wave][1] = p_conf;
        wsum[wave][2] = p_class;
    }
    __syncthreads();
    if (tid == 0) {
        float b = 0.0f, c = 0.0f, k = 0.0f;
#pragma unroll
        for (int w = 0; w < TPB / 32; ++w) {
            b += wsum[w][0];
            c += wsum[w][1];
            k += wsum[w][2];
        }
        partial[(size_t)blockIdx.x * 3 + 0] = b;
        partial[(size_t)blockIdx.x * 3 + 1] = c;
        partial[(size_t)blockIdx.x * 3 + 2] = k;
    }
}

// ---------------------------------------------------------------------------
// Kernel 2: deterministic final reduction of 1568 block partials.
__global__ __launch_bounds__(TPB) void yolo_final_kernel(
    const float* __restrict__ partial, float* __restrict__ out) {
    __shared__ float wsum[TPB / 32][3];
    const int tid = threadIdx.x;

    float b = 0.0f, c = 0.0f, k = 0.0f;
    for (int i = tid; i < NBLOCKS; i += TPB) {
        b += partial[(size_t)i * 3 + 0];
        c += partial[(size_t)i * 3 + 1];
        k += partial[(size_t)i * 3 + 2];
    }
#pragma unroll
    for (int off = 16; off >= 1; off >>= 1) {
        b += __shfl_xor(b, off, 32);
        c += __shfl_xor(c, off, 32);
        k += __shfl_xor(k, off, 32);
    }
    const int wave = tid >> 5;
    if ((tid & 31) == 0) {
        wsum[wave][0] = b;
        wsum[wave][1] = c;
        wsum[wave][2] = k;
    }
    __syncthreads();
    if (tid == 0) {
        float sb = 0.0f, sc = 0.0f, sk = 0.0f;
#pragma unroll
        for (int w = 0; w < TPB / 32; ++w) {
            sb += wsum[w][0];
            sc += wsum[w][1];
            sk += wsum[w][2];
        }
        const float inv_bs = 1.0f / (float)BATCH;
        out[0] = sb * inv_bs;   // loss_boxes
        out[1] = sc * inv_bs;   // loss_conf
        out[2] = sk * inv_bs;   // loss_class
    }
}

extern "C" void kernel_launch(void* const* d_in, const int* in_sizes, int n_in,
                              void* d_out, int out_size, void* d_ws, size_t ws_size,
                              hipStream_t stream) {
    (void)in_sizes; (void)n_in; (void)out_size; (void)ws_size;
    const float* in = (const float*)d_in[0];
    const float* tg = (const float*)d_in[1];
    float* partial = (float*)d_ws;          // NBLOCKS*3 floats
    float* out = (float*)d_out;             // 3 floats

    yolo_partial_kernel<<<NBLOCKS, TPB, 0, stream>>>(in, tg, partial);
    yolo_final_kernel<<<1, TPB, 0, stream>>>(partial, out);
}